// QuanvolutionHybrid_65481071407189
// MI455X (gfx1250) — compile-verified
//
#include <hip/hip_runtime.h>

typedef __attribute__((ext_vector_type(2))) float v2f;
typedef __attribute__((ext_vector_type(8))) float v8f;

#define KSPLIT  4                 // waves per block, each takes 1/4 of the 196 patches
#define THREADS (KSPLIT * 32)

// Fully fused: conv2x2s2 -> analytic 4-qubit circuit (<Z_w> = cumprod of
// cos(phi_c)cos(theta_c)) -> 784x10 linear via V_WMMA_F32_16X16X4_F32 ->
// log_softmax. One block = one 16-sample tile (WMMA M dim); its 4 waves
// split the K=784 reduction, combine partial C tiles through LDS, and
// split the softmax epilogue by C rows.
__global__ __launch_bounds__(THREADS)
void quanv_fused_kernel(const float* __restrict__ x,
                        const float* __restrict__ conv_w,
                        const float* __restrict__ conv_b,
                        const float* __restrict__ q_params,
                        const float* __restrict__ lin_w,
                        const float* __restrict__ lin_b,
                        float* __restrict__ out)
{
    __shared__ float red[KSPLIT][8][32];   // partial C tiles, lane-contiguous

    const int lane = threadIdx.x & 31;
    const int wave = threadIdx.x >> 5;
    const int half = lane >> 4;      // A: K pair (0,1)/(2,3); C: M rows 0..7 / 8..15
    const int m    = lane & 15;      // A: sample within tile; B/C: class N
    const int s0   = blockIdx.x * 16;

    // Uniform parameters (scalar-cached)
    const float w00 = conv_w[0],  w01 = conv_w[1],  w02 = conv_w[2],  w03 = conv_w[3];
    const float w10 = conv_w[4],  w11 = conv_w[5],  w12 = conv_w[6],  w13 = conv_w[7];
    const float w20 = conv_w[8],  w21 = conv_w[9],  w22 = conv_w[10], w23 = conv_w[11];
    const float w30 = conv_w[12], w31 = conv_w[13], w32 = conv_w[14], w33 = conv_w[15];
    const float cb0 = conv_b[0], cb1 = conv_b[1], cb2 = conv_b[2], cb3 = conv_b[3];
    const float k0 = __cosf(q_params[0] + q_params[4]);
    const float k1 = __cosf(q_params[1] + q_params[5]);
    const float k2 = __cosf(q_params[2] + q_params[6]);
    const float k3 = __cosf(q_params[3] + q_params[7]);

    // Per-lane pointers
    const float* xs = x + (size_t)(s0 + m) * 784;
    const int    nrow  = (m < 10) ? m : 9;           // keep pointer valid; mask data
    const float  bmask = (m < 10) ? 1.0f : 0.0f;     // zero-pad classes 10..15
    const float* wp = lin_w + (size_t)nrow * 784 + 2 * half;

    v8f acc = {0.f, 0.f, 0.f, 0.f, 0.f, 0.f, 0.f, 0.f};

    // Strided K-split with a fixed trip count so the unroller can work:
    // wave w handles patches p = w, w+4, ..., w+192 (49 each).
    int p = wave;
    #pragma unroll 2
    for (int i = 0; i < 49; ++i, p += KSPLIT) {
        const int pr = p / 14;
        const int pc = p - pr * 14;
        const float* xp = xs + pr * 56 + pc * 2;
        const float2 ra = *(const float2*)(xp);       // x[2pr][2pc..2pc+1]
        const float2 rb = *(const float2*)(xp + 28);  // x[2pr+1][2pc..2pc+1]

        const float t0 = fmaf(w00, ra.x, fmaf(w01, ra.y, fmaf(w02, rb.x, fmaf(w03, rb.y, cb0))));
        const float t1 = fmaf(w10, ra.x, fmaf(w11, ra.y, fmaf(w12, rb.x, fmaf(w13, rb.y, cb1))));
        const float t2 = fmaf(w20, ra.x, fmaf(w21, ra.y, fmaf(w22, rb.x, fmaf(w23, rb.y, cb2))));
        const float t3 = fmaf(w30, ra.x, fmaf(w31, ra.y, fmaf(w32, rb.x, fmaf(w33, rb.y, cb3))));

        const float z0 = k0 * __cosf(t0);
        const float z1 = k1 * __cosf(t1);
        const float z2 = k2 * __cosf(t2);
        const float z3 = k3 * __cosf(t3);

        const float e0 = z0;
        const float e1 = e0 * z1;
        const float e2 = e1 * z2;
        const float e3 = e2 * z3;

        // A fragment (16x4 f32): lanes 0-15 carry K=(0,1), lanes 16-31 K=(2,3)
        v2f A;
        A[0] = half ? e2 : e0;
        A[1] = half ? e3 : e1;

        // B fragment (4x16 f32): lane = class N, K pair selected by half
        const float2 wv = *(const float2*)(wp + 4 * p);
        v2f Bv;
        Bv[0] = wv.x * bmask;
        Bv[1] = wv.y * bmask;

        acc = __builtin_amdgcn_wmma_f32_16x16x4_f32(
            false, A, false, Bv, (short)0, acc, false, false);
    }

    // All waves publish their partial 16x16 logit tiles
    #pragma unroll
    for (int r = 0; r < 8; ++r) red[wave][r][lane] = acc[r];
    __syncthreads();

    // Epilogue split by C rows: wave w owns rows r = 2w, 2w+1.
    // C layout: VGPR r -> sample M = half*8 + r, class N = m (lanes mod 16).
    const float bias = (m < 10) ? lin_b[m] : 0.0f;
    #pragma unroll
    for (int rr = 0; rr < 2; ++rr) {
        const int r = wave * 2 + rr;
        const float v = red[0][r][lane] + red[1][r][lane] +
                        red[2][r][lane] + red[3][r][lane] + bias;
        float mx = (m < 10) ? v : -3.0e38f;
        mx = fmaxf(mx, __shfl_xor(mx, 1, 32));
        mx = fmaxf(mx, __shfl_xor(mx, 2, 32));
        mx = fmaxf(mx, __shfl_xor(mx, 4, 32));
        mx = fmaxf(mx, __shfl_xor(mx, 8, 32));
        float sm = (m < 10) ? __expf(v - mx) : 0.0f;
        sm += __shfl_xor(sm, 1, 32);
        sm += __shfl_xor(sm, 2, 32);
        sm += __shfl_xor(sm, 4, 32);
        sm += __shfl_xor(sm, 8, 32);
        if (m < 10) {
            const int sample = s0 + half * 8 + r;
            out[(size_t)sample * 10 + m] = (v - mx) - __logf(sm);
        }
    }
}

extern "C" void kernel_launch(void* const* d_in, const int* in_sizes, int n_in,
                              void* d_out, int out_size, void* d_ws, size_t ws_size,
                              hipStream_t stream) {
    (void)n_in; (void)out_size; (void)d_ws; (void)ws_size;
    const float* x        = (const float*)d_in[0];
    const float* conv_w   = (const float*)d_in[1];
    const float* conv_b   = (const float*)d_in[2];
    const float* q_params = (const float*)d_in[3];
    const float* lin_w    = (const float*)d_in[4];
    const float* lin_b    = (const float*)d_in[5];
    float* out = (float*)d_out;

    const int B = in_sizes[0] / 784;        // 8192
    const int blocks = B / 16;              // 512 blocks, one 16-sample tile each
    quanv_fused_kernel<<<dim3(blocks), dim3(THREADS), 0, stream>>>(
        x, conv_w, conv_b, q_params, lin_w, lin_b, out);
}